// MMGCNLayer_54949811585561
// MI455X (gfx1250) — compile-verified
//
#include <hip/hip_runtime.h>
#include <hip/hip_bf16.h>

#define IN_CH  256
#define OUT_CH 128

typedef __attribute__((ext_vector_type(2))) float v2f;
typedef __attribute__((ext_vector_type(8))) float v8f;

// ---------------------------------------------------------------------------
// Phase 1: xw = x @ W using V_WMMA_F32_16X16X4_F32.
// Block = 256 threads = 8 waves; block computes a 128-row x 128-col tile.
// Weight staged in LDS in fragment-ready interleaved layout:
//   lw2[p*128 + c] = (W[2p][c], W[2p+1][c])   -> B frag = one ds_load_b64.
// ---------------------------------------------------------------------------
__global__ __launch_bounds__(256) void gemm_xw_kernel(
    const float* __restrict__ x, const float* __restrict__ w,
    float* __restrict__ xw, int n_nodes) {
  extern __shared__ float lw[];  // 128 KB dynamic LDS (gfx1250 WGP = 320 KB)
  v2f* __restrict__ lw2 = (v2f*)lw;

  const int tid = threadIdx.x;

  // Stage weight into interleaved-pair layout (coalesced dword reads per row).
  #pragma unroll 4
  for (int idx = tid; idx < (IN_CH / 2) * OUT_CH; idx += 256) {
    const int p = idx >> 7;            // k-pair index 0..127
    const int c = idx & (OUT_CH - 1);  // column 0..127
    v2f v;
    v.x = w[(2 * p) * OUT_CH + c];
    v.y = w[(2 * p + 1) * OUT_CH + c];
    lw2[idx] = v;
  }
  __syncthreads();

  const int wave = tid >> 5;
  const int lane = tid & 31;
  const int half = lane >> 4;   // 0: lanes 0-15 (K=0,1)   1: lanes 16-31 (K=2,3)
  const int lid  = lane & 15;

  const long base = (long)blockIdx.x * 128 + wave * 16;  // 16-row strip start

  v8f acc[8];
  #pragma unroll
  for (int n = 0; n < 8; ++n) acc[n] = (v8f)0.f;

  // A-fragment source row for this lane (clamped; junk rows masked at store).
  long arow = base + lid;
  if (arow >= n_nodes) arow = n_nodes - 1;
  const float* aptr = x + arow * (long)IN_CH + half * 2;

  for (int k = 0; k < IN_CH; k += 4) {
    // A 16x4: lane<16 holds A[lid][k..k+1], lane>=16 holds A[lid][k+2..k+3]
    v2f a = *(const v2f*)(aptr + k);  // 8B aligned

    // B 4x16: lane<16 needs pair p=k/2, lane>=16 pair p=k/2+1, at column col.
    const v2f* bptr = lw2 + ((k >> 1) + half) * OUT_CH + lid;
    #pragma unroll
    for (int n = 0; n < 8; ++n) {
      v2f b = bptr[n * 16];           // single ds_load_b64, imm offset n*128
      acc[n] = __builtin_amdgcn_wmma_f32_16x16x4_f32(
          false, a, false, b, (short)0, acc[n], false, false);
    }
  }

  // Store: C/D layout -> VGPR r holds M=r (lanes 0-15) / M=r+8 (lanes 16-31).
  if (base + 16 <= n_nodes) {  // fast path: 781 of 782 blocks, no guards
    #pragma unroll
    for (int n = 0; n < 8; ++n) {
      const int col = n * 16 + lid;
      #pragma unroll
      for (int r = 0; r < 8; ++r)
        xw[(base + r + half * 8) * (long)OUT_CH + col] = acc[n][r];
    }
  } else {
    #pragma unroll
    for (int n = 0; n < 8; ++n) {
      const int col = n * 16 + lid;
      #pragma unroll
      for (int r = 0; r < 8; ++r) {
        const long row = base + r + half * 8;
        if (row < n_nodes) xw[row * (long)OUT_CH + col] = acc[n][r];
      }
    }
  }
}

// ---------------------------------------------------------------------------
// Phase 2: out[n, c] = bias[c]   (atomics accumulate on top of this)
// ---------------------------------------------------------------------------
__global__ __launch_bounds__(256) void init_bias_kernel(
    const float* __restrict__ bias, float* __restrict__ out, long total) {
  const long i = ((long)blockIdx.x * 256 + threadIdx.x) * 4;
  if (i < total) {
    const float4 b4 = *(const float4*)(bias + (int)(i & (OUT_CH - 1)));
    *(float4*)(out + i) = b4;
  }
}

// ---------------------------------------------------------------------------
// Phase 3: for each edge e: out[src[e], :] += xw[dst[e], :] * val[e]
// One wave per edge; lane handles 4 channels (float4 gather, 4 f32 atomics).
// Native no-return L2 atomic guaranteed via inline asm.
// ---------------------------------------------------------------------------
__device__ __forceinline__ void atomic_add_f32_native(float* p, float v) {
  asm volatile("global_atomic_add_f32 %0, %1, off scope:SCOPE_DEV"
               :: "v"(p), "v"(v) : "memory");
}

__global__ __launch_bounds__(256) void scatter_edges_kernel(
    const float* __restrict__ xw, const int* __restrict__ esrc,
    const int* __restrict__ edst, const float* __restrict__ eval,
    float* __restrict__ out, long n_edges) {
  const long t = (long)blockIdx.x * 256 + threadIdx.x;
  const long e = t >> 5;
  if (e >= n_edges) return;
  const int lane = (int)(t & 31);

  const int   d = edst[e];
  const int   s = esrc[e];
  const float v = eval[e];

  const float4 m = *(const float4*)(xw + (long)d * OUT_CH + lane * 4);
  float* o = out + (long)s * OUT_CH + lane * 4;

  atomic_add_f32_native(o + 0, m.x * v);
  atomic_add_f32_native(o + 1, m.y * v);
  atomic_add_f32_native(o + 2, m.z * v);
  atomic_add_f32_native(o + 3, m.w * v);
}

// ---------------------------------------------------------------------------
extern "C" void kernel_launch(void* const* d_in, const int* in_sizes, int n_in,
                              void* d_out, int out_size, void* d_ws, size_t ws_size,
                              hipStream_t stream) {
  const float* x    = (const float*)d_in[0];
  const int*   esrc = (const int*)  d_in[1];
  const int*   edst = (const int*)  d_in[2];
  const float* ev   = (const float*)d_in[3];
  const float* w    = (const float*)d_in[4];
  const float* bias = (const float*)d_in[5];
  float* out = (float*)d_out;
  float* xw  = (float*)d_ws;   // n_nodes * OUT_CH floats (51.2 MB)

  const int  n_nodes = in_sizes[0] / IN_CH;
  const long n_edges = in_sizes[1];

  // Phase 1: dense GEMM via WMMA (128 KB dynamic LDS for the weight)
  const int gemm_blocks = (n_nodes + 127) / 128;
  gemm_xw_kernel<<<gemm_blocks, 256, IN_CH * OUT_CH * sizeof(float), stream>>>(
      x, w, xw, n_nodes);

  // Phase 2: bias init (float4 stores)
  const long total_out = (long)n_nodes * OUT_CH;
  init_bias_kernel<<<(total_out / 4 + 255) / 256, 256, 0, stream>>>(bias, out, total_out);

  // Phase 3: edge gather/scale/scatter-add
  const long total_t = n_edges * 32;
  scatter_edges_kernel<<<(total_t + 255) / 256, 256, 0, stream>>>(
      xw, esrc, edst, ev, out, n_edges);
}